// SDConv_62242666054350
// MI455X (gfx1250) — compile-verified
//
#include <hip/hip_runtime.h>

// Problem constants (match reference)
#define N_NODES   100000
#define N_EDGES   1600000
#define K_ORDERS  3
#define CDIM      64          // IN_C == OUT_C

typedef __attribute__((ext_vector_type(2))) float v2f;
typedef __attribute__((ext_vector_type(8))) float v8f;

// -------------------------------------------------------------------------
// out[i] = bias[i & 63] for both real half [0, N*C) and imag half [N*C, 2NC)
// -------------------------------------------------------------------------
__global__ __launch_bounds__(256) void init_out_kernel(float* __restrict__ out,
                                                       const float* __restrict__ bias,
                                                       int total) {
  int i = blockIdx.x * blockDim.x + threadIdx.x;
  int stride = gridDim.x * blockDim.x;
  for (; i < total; i += stride)
    out[i] = bias[i & (CDIM - 1)];
}

// -------------------------------------------------------------------------
// YR = Xr @ Wk, YI = Xi @ Wk      (N x 64) * (64 x 64), f32, exact precision
// via V_WMMA_F32_16X16X4_F32. Block = 128 threads = 4 waves; each wave owns a
// 16x16 output tile (cols 16w..16w+15) of the block's 16-row stripe.
//
// A fragment (16x4 f32, 2 VGPRs/lane): lane l holds row M=(l&15),
//   K = {0,1} for lanes 0-15, K = {2,3} for lanes 16-31 (ISA 7.12.2).
// B fragment (4x16 f32): lane l holds col N=(l&15), same K split as A.
// C/D (16x16 f32, 8 VGPRs): VGPR v -> M = v (lanes 0-15) / v+8 (lanes 16-31),
//   N = lane&15.
// -------------------------------------------------------------------------
__global__ __launch_bounds__(128) void gemm_xw_kernel(const float* __restrict__ Xr,
                                                      const float* __restrict__ Xi,
                                                      const float* __restrict__ Wk,
                                                      float* __restrict__ YR,
                                                      float* __restrict__ YI) {
  const int lane    = threadIdx.x & 31;
  const int wave    = threadIdx.x >> 5;
  const int rowBase = blockIdx.x << 4;       // 16 rows per block (N/16 = 6250 blocks)
  const int colBase = wave << 4;             // 16 cols per wave

  const int mrow = rowBase + (lane & 15);
  const int hi   = lane >> 4;                // half-wave selector
  const int koff = hi * 2;
  const int ncol = lane & 15;

  const float* __restrict__ arp = Xr + (size_t)mrow * CDIM;
  const float* __restrict__ aip = Xi + (size_t)mrow * CDIM;

  v8f cr = {0.f, 0.f, 0.f, 0.f, 0.f, 0.f, 0.f, 0.f};
  v8f ci = {0.f, 0.f, 0.f, 0.f, 0.f, 0.f, 0.f, 0.f};

#pragma unroll
  for (int s = 0; s < CDIM / 4; ++s) {       // 16 K-steps of 4 -> 32 WMMAs total
    const int k0 = s * 4 + koff;             // this lane's K pair
    v2f ar = *reinterpret_cast<const v2f*>(arp + k0);   // global_load_b64
    v2f ai = *reinterpret_cast<const v2f*>(aip + k0);
    v2f b;
    b.x = Wk[(k0)     * CDIM + colBase + ncol];
    b.y = Wk[(k0 + 1) * CDIM + colBase + ncol];
    cr = __builtin_amdgcn_wmma_f32_16x16x4_f32(false, ar, false, b, (short)0, cr,
                                               false, false);
    ci = __builtin_amdgcn_wmma_f32_16x16x4_f32(false, ai, false, b, (short)0, ci,
                                               false, false);
  }

#pragma unroll
  for (int v = 0; v < 8; ++v) {
    const int r = rowBase + v + hi * 8;
    const int c = colBase + ncol;
    YR[(size_t)r * CDIM + c] = cr[v];
    YI[(size_t)r * CDIM + c] = ci[v];
  }
}

// -------------------------------------------------------------------------
// Edge scatter for one spectral order k:
//   outR[r,:] += lr * YR[c,:] - li * YI[c,:]
//   outI[r,:] += li * YR[c,:] + lr * YI[c,:]
// One wave per edge; each lane owns 2 channels (float2). Gathers of Y and the
// f32 atomics both land in the 192MB L2 (Y: 51MB, out: 51MB resident).
// -------------------------------------------------------------------------
__global__ __launch_bounds__(256) void scatter_kernel(const int* __restrict__ rows,
                                                      const int* __restrict__ cols,
                                                      const float* __restrict__ Lr,
                                                      const float* __restrict__ Li,
                                                      const float* __restrict__ YR,
                                                      const float* __restrict__ YI,
                                                      float* __restrict__ outR,
                                                      float* __restrict__ outI) {
  const int lane = threadIdx.x & 31;
  const int ch   = lane * 2;
  int e = (int)((blockIdx.x * (unsigned)blockDim.x + threadIdx.x) >> 5);
  const int nwaves = (int)((gridDim.x * (unsigned)blockDim.x) >> 5);

  for (; e < N_EDGES; e += nwaves) {
    const int   r  = rows[e];
    const int   c  = cols[e];
    const float lr = Lr[e];
    const float li = Li[e];

    v2f yr = *reinterpret_cast<const v2f*>(YR + (size_t)c * CDIM + ch);
    v2f yi = *reinterpret_cast<const v2f*>(YI + (size_t)c * CDIM + ch);

    const float trx = lr * yr.x - li * yi.x;
    const float try_ = lr * yr.y - li * yi.y;
    const float tix = li * yr.x + lr * yi.x;
    const float tiy = li * yr.y + lr * yi.y;

    float* pr = outR + (size_t)r * CDIM + ch;
    float* pi = outI + (size_t)r * CDIM + ch;
    unsafeAtomicAdd(pr,     trx);   // global_atomic_add_f32 (L2-resident)
    unsafeAtomicAdd(pr + 1, try_);
    unsafeAtomicAdd(pi,     tix);
    unsafeAtomicAdd(pi + 1, tiy);
  }
}

// -------------------------------------------------------------------------
// Launch: init out with bias, then for each k: GEMM (WMMA f32) -> scatter.
// d_ws holds YR|YI for the current k only (2 * N * 64 f32 = 51.2 MB).
// -------------------------------------------------------------------------
extern "C" void kernel_launch(void* const* d_in, const int* in_sizes, int n_in,
                              void* d_out, int out_size, void* d_ws, size_t ws_size,
                              hipStream_t stream) {
  const float* Xr    = (const float*)d_in[0];      // (N, 64)
  const float* Xi    = (const float*)d_in[1];      // (N, 64)
  const int*   eidx  = (const int*)  d_in[2];      // (2, E): rows then cols
  const float* LrAll = (const float*)d_in[3];      // (K, E)
  const float* LiAll = (const float*)d_in[4];      // (K, E)
  const float* W     = (const float*)d_in[5];      // (K, 64, 64)
  const float* bias  = (const float*)d_in[6];      // (1, 64)

  float* outR = (float*)d_out;                     // (N, 64)
  float* outI = outR + (size_t)N_NODES * CDIM;     // (N, 64)

  float* YR = (float*)d_ws;                        // (N, 64) scratch, reused per k
  float* YI = YR + (size_t)N_NODES * CDIM;

  const int* rows = eidx;
  const int* cols = eidx + N_EDGES;

  // 1) out = bias (broadcast); makes the atomic accumulation land on bias.
  {
    const int total = 2 * N_NODES * CDIM;
    init_out_kernel<<<4096, 256, 0, stream>>>(outR, bias, total);
  }

  // 2) per spectral order: dense projection first (WMMA f32), then scatter.
  for (int k = 0; k < K_ORDERS; ++k) {
    gemm_xw_kernel<<<N_NODES / 16, 128, 0, stream>>>(
        Xr, Xi, W + (size_t)k * CDIM * CDIM, YR, YI);

    scatter_kernel<<<N_EDGES / 8, 256, 0, stream>>>(
        rows, cols, LrAll + (size_t)k * N_EDGES, LiAll + (size_t)k * N_EDGES,
        YR, YI, outR, outI);
  }
}